// MessagePassingLayer_34832184770731
// MI455X (gfx1250) — compile-verified
//
#include <hip/hip_runtime.h>

typedef __attribute__((ext_vector_type(2))) float v2f;
typedef __attribute__((ext_vector_type(8))) float v8f;

#define IN_DIM   128
#define OUT_DIM  128
#define EDGE_DIM 64

// ---------------------------------------------------------------------------
// Kernel 1: project node table once.
//   hu = h @ W_hu^T   [N, OUT]
//   hw = h @ W_hw^T   [N, OUT]
// One wave computes one 16-row M-tile across all 8 N-tiles for BOTH weight
// matrices, reusing the preloaded A fragments of h.
// WMMA f32 16x16x4 layouts (ISA 7.12.2):
//   A (16x4):  lane m=lane&15, vgpr{0,1} = h[m][k0+2*hi], h[m][k0+2*hi+1]
//   B (4x16):  lane n=lane&15, vgpr{0,1} = W[n][k0+2*hi], W[n][k0+2*hi+1]
//   C/D:       vgpr r -> row r+8*hi, col lane&15
// ---------------------------------------------------------------------------
__global__ __launch_bounds__(256)
void mp_node_project(const float* __restrict__ h,
                     const float* __restrict__ W_hu,
                     const float* __restrict__ W_hw,
                     float* __restrict__ hu,
                     float* __restrict__ hw,
                     int numTilesM)
{
    const int wave = (int)((blockIdx.x * blockDim.x + threadIdx.x) >> 5);
    if (wave >= numTilesM) return;          // wave-uniform: EXEC stays all-1s
    const int lane = (int)(threadIdx.x & 31);
    const int m    = lane & 15;
    const int hi   = lane >> 4;
    const int mBase = wave * 16;

    // Preload A fragments of h tile: 32 K-blocks of float2 (64 VGPRs)
    v2f a[IN_DIM / 4];
    const float* hrow = h + (size_t)(mBase + m) * IN_DIM + 2 * hi;
#pragma unroll
    for (int kb = 0; kb < IN_DIM / 4; ++kb)
        a[kb] = *(const v2f*)(hrow + kb * 4);

#pragma unroll
    for (int nt = 0; nt < OUT_DIM / 16; ++nt) {
        const int nBase = nt * 16;
        const float* wu = W_hu + (size_t)(nBase + m) * IN_DIM + 2 * hi;
        const float* ww = W_hw + (size_t)(nBase + m) * IN_DIM + 2 * hi;
        v8f accU = {};
        v8f accW = {};
#pragma unroll
        for (int kb = 0; kb < IN_DIM / 4; ++kb) {
            v2f bu = *(const v2f*)(wu + kb * 4);
            v2f bw = *(const v2f*)(ww + kb * 4);
            accU = __builtin_amdgcn_wmma_f32_16x16x4_f32(
                false, a[kb], false, bu, (short)0, accU, false, false);
            accW = __builtin_amdgcn_wmma_f32_16x16x4_f32(
                false, a[kb], false, bw, (short)0, accW, false, false);
        }
#pragma unroll
        for (int r = 0; r < 8; ++r) {
            const size_t o = (size_t)(mBase + r + 8 * hi) * OUT_DIM + nBase + m;
            hu[o] = accU[r];
            hw[o] = accW[r];
        }
    }
}

// ---------------------------------------------------------------------------
// Kernel 2: fused edge kernel.
//   out[e] = e_feat[e] @ W_e^T  +  hu[src[e]]  +  hw[tgt[e]]
// One wave = one 16-edge tile. Accumulator is INITIALIZED with the gathered
// hu/hw rows (tables are 51 MB total -> resident in 192 MB L2), then the
// edge-feature GEMM (K=64) is WMMA-accumulated on top. Single pass over out.
// ---------------------------------------------------------------------------
__global__ __launch_bounds__(256)
void mp_edge_message(const float* __restrict__ efeat,       // [E, EDGE_DIM]
                     const long long* __restrict__ edge_index, // [2, E] int64
                     const float* __restrict__ W_e,          // [OUT, EDGE_DIM]
                     const float* __restrict__ hu,           // [N, OUT]
                     const float* __restrict__ hw,           // [N, OUT]
                     float* __restrict__ out,                // [E, OUT]
                     int numTilesE, int E)
{
    const int wave = (int)((blockIdx.x * blockDim.x + threadIdx.x) >> 5);
    if (wave >= numTilesE) return;          // wave-uniform
    const int lane = (int)(threadIdx.x & 31);
    const int m    = lane & 15;
    const int hi   = lane >> 4;
    const int eBase = wave * 16;

    // Row indices for the 8 output rows this lane's accumulator covers
    long long srcIdx[8], tgtIdx[8];
#pragma unroll
    for (int r = 0; r < 8; ++r) {
        const int erow = eBase + r + 8 * hi;
        srcIdx[r] = edge_index[erow];
        tgtIdx[r] = edge_index[(size_t)E + erow];
    }

    // Preload A fragments of the 16x64 edge-feature tile (16 K-blocks)
    v2f a[EDGE_DIM / 4];
    const float* erow = efeat + (size_t)(eBase + m) * EDGE_DIM + 2 * hi;
#pragma unroll
    for (int kb = 0; kb < EDGE_DIM / 4; ++kb)
        a[kb] = *(const v2f*)(erow + kb * 4);

#pragma unroll
    for (int nt = 0; nt < OUT_DIM / 16; ++nt) {
        const int n = nt * 16 + m;

        // Init accumulator with L2-resident gathers: hu[src] + hw[tgt].
        // For fixed r, lanes 0-15 / 16-31 each read a contiguous 64B segment.
        v8f acc;
#pragma unroll
        for (int r = 0; r < 8; ++r)
            acc[r] = hu[(size_t)srcIdx[r] * OUT_DIM + n] +
                     hw[(size_t)tgtIdx[r] * OUT_DIM + n];

        // ee tile: K = 64 -> 16 WMMA f32 16x16x4
        const float* wb = W_e + (size_t)n * EDGE_DIM + 2 * hi;
#pragma unroll
        for (int kb = 0; kb < EDGE_DIM / 4; ++kb) {
            v2f b = *(const v2f*)(wb + kb * 4);
            acc = __builtin_amdgcn_wmma_f32_16x16x4_f32(
                false, a[kb], false, b, (short)0, acc, false, false);
        }

#pragma unroll
        for (int r = 0; r < 8; ++r)
            out[(size_t)(eBase + r + 8 * hi) * OUT_DIM + n] = acc[r];
    }
}

extern "C" void kernel_launch(void* const* d_in, const int* in_sizes, int n_in,
                              void* d_out, int out_size, void* d_ws, size_t ws_size,
                              hipStream_t stream) {
    const float*     h          = (const float*)d_in[0];      // [N, 128]
    const float*     efeat      = (const float*)d_in[1];      // [E, 64]
    const long long* edge_index = (const long long*)d_in[2];  // [2, E] int64
    const float*     W_e        = (const float*)d_in[3];      // [128, 64]
    const float*     W_hu       = (const float*)d_in[4];      // [128, 128]
    const float*     W_hw       = (const float*)d_in[5];      // [128, 128]
    float*           out        = (float*)d_out;              // [E, 128]

    const int N = in_sizes[0] / IN_DIM;    // 50000
    const int E = in_sizes[1] / EDGE_DIM;  // 800000

    // Workspace: projected node tables hu, hw  (2 * N * 128 * 4B = 51.2 MB)
    float* hu = (float*)d_ws;
    float* hw = hu + (size_t)N * OUT_DIM;

    const int tilesM = N / 16;             // dims divide exactly (3125)
    const int tilesE = E / 16;             // 50000

    const int blocksA = (tilesM + 7) / 8;  // 8 waves (256 thr) per block
    mp_node_project<<<blocksA, 256, 0, stream>>>(h, W_hu, W_hw, hu, hw, tilesM);

    const int blocksB = (tilesE + 7) / 8;
    mp_edge_message<<<blocksB, 256, 0, stream>>>(efeat, edge_index, W_e,
                                                 hu, hw, out, tilesE, E);
}